// LinkPrediction_82394652607278
// MI455X (gfx1250) — compile-verified
//
#include <hip/hip_runtime.h>
#include <hip/hip_bf16.h>

// Shapes from the reference
#define IN_CH  128
#define HID_CH 128
#define OUT_CH 64

typedef __attribute__((ext_vector_type(2))) float v2f;
typedef __attribute__((ext_vector_type(8))) float v8f;

// ---------------------------------------------------------------------------
// degree helpers
// ---------------------------------------------------------------------------
__global__ __launch_bounds__(256) void fill_f32(float* __restrict__ p, float v, int n) {
  int i = blockIdx.x * blockDim.x + threadIdx.x;
  if (i < n) p[i] = v;
}

__global__ __launch_bounds__(256) void deg_count(const int* __restrict__ col,
                                                 float* __restrict__ deg, int e) {
  int i = blockIdx.x * blockDim.x + threadIdx.x;
  if (i < e) atomicAdd(&deg[col[i]], 1.0f);
}

__global__ __launch_bounds__(256) void deg_finalize(const float* __restrict__ deg,
                                                    float* __restrict__ dis,
                                                    float* __restrict__ dinv, int n) {
  int i = blockIdx.x * blockDim.x + threadIdx.x;
  if (i < n) {
    float d = deg[i];
    dis[i]  = rsqrtf(d);
    dinv[i] = 1.0f / d;
  }
}

// ---------------------------------------------------------------------------
// Dense GEMM C[M x NDIM] = A[M x KDIM] @ W[KDIM x NDIM] via V_WMMA_F32_16X16X4_F32
// One wave computes one 16x16 tile; blockDim.x = 32 * (NDIM/16) so the waves of
// a block cover one full 16-row stripe. K advances 4 per WMMA.
// A-frag (16x4, 32-bit): lanes 0-15 carry K=0,1 ; lanes 16-31 carry K=2,3.
// B-frag (4x16): row K striped across lanes within a VGPR.
// C/D: VGPR v -> M = v + 8*(lane/16), N = lane&15.
// EXEC must be all-1s for WMMA: clamp addresses (uniform fast path), never
// predicate the accumulate loop. Store guard hoisted to ONE uniform branch so
// full tiles (the only kind when M % 16 == 0) emit 8 straight-line b32 stores.
// ---------------------------------------------------------------------------
template <int KDIM, int NDIM>
__global__ __launch_bounds__(32 * (NDIM / 16)) void gemm_wmma_f32(
    const float* __restrict__ A, const float* __restrict__ W,
    float* __restrict__ C, int M) {
  const int lane = threadIdx.x & 31;
  const int wave = threadIdx.x >> 5;            // tile column index
  const int tm   = blockIdx.x;                  // tile row index
  const int colT = (wave << 4) + (lane & 15);   // global output column
  const int half = (lane >> 4) << 1;            // 0 (lanes 0-15) or 2 (lanes 16-31)

  const bool fullTile = ((tm << 4) + 16) <= M;  // uniform across the wave
  int row = (tm << 4) + (lane & 15);
  if (!fullTile && row >= M) row = M - 1;       // clamp only on the tail tile

  const float* arow = A + (size_t)row * KDIM + half;

  v8f acc = {};
#pragma unroll 8
  for (int k = 0; k < KDIM; k += 4) {
    float2 av = *(const float2*)(arow + k);     // contiguous 8B load
    v2f a, b;
    a.x = av.x;
    a.y = av.y;
    b.x = W[(size_t)(k + half) * NDIM + colT];
    b.y = W[(size_t)(k + half + 1) * NDIM + colT];
    acc = __builtin_amdgcn_wmma_f32_16x16x4_f32(false, a, false, b,
                                                (short)0, acc, false, false);
  }

  const int rbase = (tm << 4) + ((lane >> 4) << 3);
  float* cp = C + (size_t)rbase * NDIM + colT;
  if (fullTile) {
#pragma unroll
    for (int v = 0; v < 8; ++v) cp[(size_t)v * NDIM] = acc[v];
  } else {
#pragma unroll
    for (int v = 0; v < 8; ++v) {
      if (rbase + v < M) cp[(size_t)v * NDIM] = acc[v];
    }
  }
}

// ---------------------------------------------------------------------------
// Edge aggregation: agg[col] += h[row] * dis[row]*dis[col]
// One wave per edge; lane handles C/32 contiguous channels (float4 / float2
// gather, per-float global_atomic_add_f32 scatter -> all L2-resident).
// ---------------------------------------------------------------------------
template <int C>
__global__ __launch_bounds__(256) void scatter_agg(
    const float* __restrict__ h, const int* __restrict__ row,
    const int* __restrict__ col, const float* __restrict__ dis,
    float* __restrict__ agg, int nE) {
  constexpr int VEC = C / 32;
  typedef float vldT __attribute__((ext_vector_type(VEC)));

  int e = blockIdx.x * (blockDim.x >> 5) + (threadIdx.x >> 5);
  if (e >= nE) return;
  int lane = threadIdx.x & 31;

  int r = row[e];
  int c = col[e];
  float norm = dis[r] * dis[c];

  vldT hv = *(const vldT*)(h + (size_t)r * C + lane * VEC);
  float* ad = agg + (size_t)c * C + lane * VEC;
#pragma unroll
  for (int i = 0; i < VEC; ++i) atomicAdd(ad + i, hv[i] * norm);
}

// out = agg + h * deg_inv + bias (optional ReLU), written in place into agg
template <int C, bool RELU>
__global__ __launch_bounds__(256) void combine(
    float* __restrict__ agg, const float* __restrict__ h,
    const float* __restrict__ dinv, const float* __restrict__ bias,
    size_t total) {
  size_t i = (size_t)blockIdx.x * blockDim.x + threadIdx.x;
  if (i >= total) return;
  int node = (int)(i / C);
  int ch   = (int)(i % C);
  float v = agg[i] + h[i] * dinv[node] + bias[ch];
  if (RELU) v = fmaxf(v, 0.0f);
  agg[i] = v;
}

// ---------------------------------------------------------------------------
// Decode: out[e] = dot(z[src[e]], z[dst[e]]) over OUT_CH=64 channels.
// One wave per labeled edge: lane holds a float2, wave32 shfl_xor reduce.
// ---------------------------------------------------------------------------
__global__ __launch_bounds__(256) void decode_dot(
    const float* __restrict__ z, const int* __restrict__ src,
    const int* __restrict__ dst, float* __restrict__ out, int nL) {
  int e = blockIdx.x * (blockDim.x >> 5) + (threadIdx.x >> 5);
  if (e >= nL) return;
  int lane = threadIdx.x & 31;

  int s = src[e];
  int d = dst[e];
  float2 a = *(const float2*)(z + (size_t)s * OUT_CH + lane * 2);
  float2 b = *(const float2*)(z + (size_t)d * OUT_CH + lane * 2);
  float p = a.x * b.x + a.y * b.y;
#pragma unroll
  for (int off = 16; off > 0; off >>= 1) p += __shfl_xor(p, off, 32);
  if (lane == 0) out[e] = p;
}

// ---------------------------------------------------------------------------
extern "C" void kernel_launch(void* const* d_in, const int* in_sizes, int n_in,
                              void* d_out, int out_size, void* d_ws, size_t ws_size,
                              hipStream_t stream) {
  const float* x   = (const float*)d_in[0];
  const float* W1  = (const float*)d_in[1];
  const float* b1  = (const float*)d_in[2];
  const float* W2  = (const float*)d_in[3];
  const float* b2  = (const float*)d_in[4];
  const int*  eidx = (const int*)d_in[5];   // [2, E] flat: first E = row, next E = col
  const int*  lidx = (const int*)d_in[6];   // [2, L] flat

  const int N = in_sizes[0] / IN_CH;
  const int E = in_sizes[5] / 2;
  const int L = in_sizes[6] / 2;

  const int* e_row = eidx;
  const int* e_col = eidx + E;
  const int* l_src = lidx;
  const int* l_dst = lidx + L;

  // Carve workspace (256B aligned chunks)
  size_t off = 0;
  auto alloc = [&](size_t bytes) -> float* {
    float* p = (float*)((char*)d_ws + off);
    off += (bytes + 255) & ~(size_t)255;
    return p;
  };
  float* deg  = alloc((size_t)N * sizeof(float));
  float* dis  = alloc((size_t)N * sizeof(float));
  float* dinv = alloc((size_t)N * sizeof(float));
  float* h1   = alloc((size_t)N * HID_CH * sizeof(float));
  float* agg1 = alloc((size_t)N * HID_CH * sizeof(float));
  float* h2   = alloc((size_t)N * OUT_CH * sizeof(float));
  float* agg2 = alloc((size_t)N * OUT_CH * sizeof(float));

  // Re-zero accumulators every call (graph-capture-safe memset nodes)
  hipMemsetAsync(agg1, 0, (size_t)N * HID_CH * sizeof(float), stream);
  hipMemsetAsync(agg2, 0, (size_t)N * OUT_CH * sizeof(float), stream);

  // Degrees (self-loop => start at 1)
  fill_f32<<<(N + 255) / 256, 256, 0, stream>>>(deg, 1.0f, N);
  deg_count<<<(E + 255) / 256, 256, 0, stream>>>(e_col, deg, E);
  deg_finalize<<<(N + 255) / 256, 256, 0, stream>>>(deg, dis, dinv, N);

  const int mtiles = (N + 15) / 16;

  // Layer 1: h1 = x @ W1 ; aggregate ; agg1 = relu(agg1 + h1*dinv + b1)
  gemm_wmma_f32<IN_CH, HID_CH>
      <<<mtiles, 32 * (HID_CH / 16), 0, stream>>>(x, W1, h1, N);
  scatter_agg<HID_CH>
      <<<(E + 7) / 8, 256, 0, stream>>>(h1, e_row, e_col, dis, agg1, E);
  {
    size_t total = (size_t)N * HID_CH;
    combine<HID_CH, true>
        <<<(unsigned)((total + 255) / 256), 256, 0, stream>>>(agg1, h1, dinv, b1, total);
  }

  // Layer 2: h2 = agg1 @ W2 ; aggregate ; agg2 = agg2 + h2*dinv + b2
  gemm_wmma_f32<HID_CH, OUT_CH>
      <<<mtiles, 32 * (OUT_CH / 16), 0, stream>>>(agg1, W2, h2, N);
  scatter_agg<OUT_CH>
      <<<(E + 7) / 8, 256, 0, stream>>>(h2, e_row, e_col, dis, agg2, E);
  {
    size_t total = (size_t)N * OUT_CH;
    combine<OUT_CH, false>
        <<<(unsigned)((total + 255) / 256), 256, 0, stream>>>(agg2, h2, dinv, b2, total);
  }

  // Decode
  decode_dot<<<(L + 7) / 8, 256, 0, stream>>>(agg2, l_src, l_dst, (float*)d_out, L);
}